// IMEGPT2Attention_14637248545507
// MI455X (gfx1250) — compile-verified
//
#include <hip/hip_runtime.h>
#include <hip/hip_bf16.h>
#include <cstdint>

typedef __attribute__((ext_vector_type(2))) float v2f;
typedef __attribute__((ext_vector_type(8))) float v8f;

// D = A(16x4) * B(4x16) + C, fp32 matrix core
#define WMMA4(a, b, c) \
  __builtin_amdgcn_wmma_f32_16x16x4_f32(false, (a), false, (b), (short)0, (c), false, false)

constexpr int Bb = 2, Sq = 2048, Dm = 1024, Hh = 16, HD = 64;
constexpr int BS = Bb * Sq;             // 4096 rows
constexpr float NEG_BIG = -1e30f;

// Async global -> LDS copy (16B per lane), tracked by ASYNCcnt.
// GVS mode: mem = SGPR_base + VGPR_byte_offset ; LDS dest = low 32 bits of
// the generic shared pointer (LDS aperture truncation per ISA 10.2).
__device__ __forceinline__ void async_b128(uint32_t lds_addr, uint32_t byte_off,
                                           const float* base) {
  asm volatile("global_load_async_to_lds_b128 %0, %1, %2"
               :: "v"(lds_addr), "v"(byte_off), "s"(base)
               : "memory");
}
__device__ __forceinline__ void wait_async0() {
  asm volatile("s_wait_asynccnt 0x0" ::: "memory");
}
__device__ __forceinline__ uint32_t lds_u32(const void* p) {
  return (uint32_t)(uintptr_t)p;
}

// ---------------------------------------------------------------------------
// GEMM: out[M=BS x NCOLS] = X[BS x 1024] * W[1024 x NCOLS] + bias
// block = 128 threads (4 waves), computes 16 rows x 64 cols.
// Double-buffered async LDS staging of A (16x64) and B (64x64) tiles.
// MODE 0: scatter to q/k/v head layout [B,H,S,HD].  MODE 1: plain row-major.
// ---------------------------------------------------------------------------
template <int NCOLS, int MODE>
__global__ __launch_bounds__(128) void gemm_kernel(
    const float* __restrict__ X, const float* __restrict__ W,
    const float* __restrict__ bias, float* __restrict__ out0,
    float* __restrict__ out1, float* __restrict__ out2) {
  __shared__ float lA[2][16 * 64];
  __shared__ float lB[2][64 * 64];

  const int tid  = threadIdx.x;
  const int lane = tid & 31;
  const int wv   = tid >> 5;
  const int n    = lane & 15;       // N (and A-row M) index
  const int hi   = lane >> 4;       // lane half
  const int k0   = hi * 2;          // K sub-pair selector
  const int row0 = blockIdx.y * 16;
  const int col0 = blockIdx.x * 64;
  const int col  = col0 + wv * 16 + n;

  v8f acc;
  {
    float bv = bias[col];
#pragma unroll
    for (int i = 0; i < 8; ++i) acc[i] = bv;
  }

  auto stage = [&](int kk0, int buf) {
    // A tile: 1024 floats, 8 per thread (two b128s, same row)
    {
      uint32_t ia = (uint32_t)tid * 8;
      uint32_t r = ia >> 6, c = ia & 63;
      uint32_t goff = (((uint32_t)(row0 + r)) * Dm + kk0 + c) * 4u;
      uint32_t la = lds_u32(&lA[buf][ia]);
      async_b128(la, goff, X);
      async_b128(la + 16u, goff + 16u, X);
    }
    // B tile: 4096 floats, 32 per thread (eight b128s)
    uint32_t lb0 = lds_u32(&lB[buf][0]);
#pragma unroll
    for (int p = 0; p < 8; ++p) {
      uint32_t idx = (uint32_t)p * 512 + (uint32_t)tid * 4;
      uint32_t r = idx >> 6, c = idx & 63;
      uint32_t goff = (((uint32_t)(kk0 + r)) * NCOLS + col0 + c) * 4u;
      async_b128(lb0 + idx * 4u, goff, W);
    }
  };

  stage(0, 0);
  wait_async0();
  __syncthreads();

  int it = 0;
  for (int kk0 = 0; kk0 < Dm; kk0 += 64, ++it) {
    const int cur = it & 1;
    if (kk0 + 64 < Dm) stage(kk0 + 64, 1 - cur);   // prefetch next tile
#pragma unroll
    for (int j = 0; j < 16; ++j) {
      v2f a = *(const v2f*)&lA[cur][n * 64 + 4 * j + k0];
      v2f b;
      b.x = lB[cur][(4 * j + k0) * 64 + wv * 16 + n];
      b.y = lB[cur][(4 * j + k0 + 1) * 64 + wv * 16 + n];
      acc = WMMA4(a, b, acc);
    }
    wait_async0();
    __syncthreads();
  }

#pragma unroll
  for (int i = 0; i < 8; ++i) {
    int row = row0 + hi * 8 + i;
    if constexpr (MODE == 0) {
      int which = col >> 10;           // 0:q 1:k 2:v (uniform per block)
      int d = col & 1023;
      int h = d >> 6, hd = d & 63;
      int b = row >> 11, s = row & 2047;
      float* dst = (which == 0) ? out0 : (which == 1 ? out1 : out2);
      dst[(((size_t)b * Hh + h) * Sq + s) * HD + hd] = acc[i];
    } else {
      out0[(size_t)row * NCOLS + col] = acc[i];
    }
  }
}

// ---------------------------------------------------------------------------
// Causal flash attention. Q,K,V in [B,H,S,HD] fp32. Output to [B,S,D].
// grid = (S/64, B*H), block = 128 (4 waves, each wave owns 16 q rows).
// K/V chunks (64 keys x 64 dims) double-buffered via async LDS copies.
// ---------------------------------------------------------------------------
__global__ __launch_bounds__(128) void attn_kernel(
    const float* __restrict__ Q, const float* __restrict__ Kd,
    const float* __restrict__ V, float* __restrict__ out) {
  __shared__ float lK[2][64 * 64];
  __shared__ float lV[2][64 * 64];
  __shared__ float lP[4 * 16 * 16];

  const int tid  = threadIdx.x;
  const int lane = tid & 31;
  const int wv   = tid >> 5;
  const int n    = lane & 15;
  const int hi   = lane >> 4;
  const int k0   = hi * 2;
  const int bh    = blockIdx.y;              // b*H + h
  const int qBase = blockIdx.x * 64;
  const int qglob = qBase + wv * 16;         // first q row of this wave

  // Q A-fragments: rows qglob..qglob+15, full HD=64 (16 K-steps of 4)
  const float* qp = Q + ((size_t)bh * Sq + qglob) * HD;
  v2f aq[16];
#pragma unroll
  for (int j = 0; j < 16; ++j)
    aq[j] = *(const v2f*)&qp[(size_t)n * HD + 4 * j + k0];

  v8f o0, o1, o2, o3;
  float mrow[8], lrow[8];
#pragma unroll
  for (int i = 0; i < 8; ++i) {
    o0[i] = o1[i] = o2[i] = o3[i] = 0.f;
    mrow[i] = NEG_BIG;
    lrow[i] = 0.f;
  }

  const float* kbase = Kd + (size_t)bh * Sq * HD;
  const float* vbase = V  + (size_t)bh * Sq * HD;

  auto stage = [&](int kc, int buf) {
    uint32_t lk0 = lds_u32(&lK[buf][0]);
    uint32_t lv0 = lds_u32(&lV[buf][0]);
    uint32_t cbase = (uint32_t)kc * 64u * HD * 4u;       // chunk byte offset
#pragma unroll
    for (int p = 0; p < 8; ++p) {
      uint32_t idx  = (uint32_t)p * 512 + (uint32_t)tid * 4;
      uint32_t boff = idx * 4u;
      async_b128(lk0 + boff, cbase + boff, kbase);
      async_b128(lv0 + boff, cbase + boff, vbase);
    }
  };

  const int nChunks = qBase / 64 + 1;        // causal: keys 0..qBase+63
  stage(0, 0);
  wait_async0();
  __syncthreads();

  for (int kc = 0; kc < nChunks; ++kc) {
    const int cur = kc & 1;
    const int kBase = kc * 64;
    if (kc + 1 < nChunks) stage(kc + 1, 1 - cur);        // prefetch next chunk

#pragma unroll
    for (int st = 0; st < 4; ++st) {
      const int keyBase = kBase + st * 16;
      // wave-uniform predicate, forced scalar so EXEC stays all-1s at WMMA
      if (__builtin_amdgcn_readfirstlane(keyBase <= qglob + 15 ? 1 : 0)) {
        // ---- scores = Q * K^T (16x16) ----
        v8f sc;
#pragma unroll
        for (int i = 0; i < 8; ++i) sc[i] = 0.f;
#pragma unroll
        for (int j = 0; j < 16; ++j) {
          v2f b = *(const v2f*)&lK[cur][(st * 16 + n) * 64 + 4 * j + k0];
          sc = WMMA4(aq[j], b, sc);
        }

        // ---- causal mask + online softmax (row spread across lanes n) ----
        float pr[8], scl[8];
#pragma unroll
        for (int i = 0; i < 8; ++i) {
          int row = qglob + hi * 8 + i;
          float s = sc[i];
          if (keyBase + n > row) s = NEG_BIG;
          float r = s;
          r = fmaxf(r, __shfl_xor(r, 1, 32));
          r = fmaxf(r, __shfl_xor(r, 2, 32));
          r = fmaxf(r, __shfl_xor(r, 4, 32));
          r = fmaxf(r, __shfl_xor(r, 8, 32));
          float mnew = fmaxf(mrow[i], r);
          float cold = __expf(mrow[i] - mnew);
          float p = __expf(s - mnew);
          float ps = p;
          ps += __shfl_xor(ps, 1, 32);
          ps += __shfl_xor(ps, 2, 32);
          ps += __shfl_xor(ps, 4, 32);
          ps += __shfl_xor(ps, 8, 32);
          lrow[i] = lrow[i] * cold + ps;
          mrow[i] = mnew;
          pr[i] = p;
          scl[i] = cold;
        }
#pragma unroll
        for (int i = 0; i < 8; ++i) {
          o0[i] *= scl[i]; o1[i] *= scl[i]; o2[i] *= scl[i]; o3[i] *= scl[i];
        }

        // ---- P (D-layout) -> LDS -> A-layout, then O += P * V ----
#pragma unroll
        for (int i = 0; i < 8; ++i)
          lP[wv * 256 + (hi * 8 + i) * 16 + n] = pr[i];
#pragma unroll
        for (int j = 0; j < 4; ++j) {
          v2f a = *(const v2f*)&lP[wv * 256 + n * 16 + 4 * j + k0];
          int kr = st * 16 + 4 * j + k0;
          v2f b;
          b.x = lV[cur][(kr) * 64 +  0 + n]; b.y = lV[cur][(kr + 1) * 64 +  0 + n];
          o0 = WMMA4(a, b, o0);
          b.x = lV[cur][(kr) * 64 + 16 + n]; b.y = lV[cur][(kr + 1) * 64 + 16 + n];
          o1 = WMMA4(a, b, o1);
          b.x = lV[cur][(kr) * 64 + 32 + n]; b.y = lV[cur][(kr + 1) * 64 + 32 + n];
          o2 = WMMA4(a, b, o2);
          b.x = lV[cur][(kr) * 64 + 48 + n]; b.y = lV[cur][(kr + 1) * 64 + 48 + n];
          o3 = WMMA4(a, b, o3);
        }
      }
    }
    wait_async0();
    __syncthreads();
  }

  // epilogue: normalize rows and merge heads into [B,S,D]
  const int b = bh / Hh, h = bh % Hh;
#pragma unroll
  for (int i = 0; i < 8; ++i) {
    float inv = 1.0f / lrow[i];
    int row = qglob + hi * 8 + i;
    float* dst = out + ((size_t)b * Sq + row) * Dm + h * HD;
    dst[ 0 + n] = o0[i] * inv;
    dst[16 + n] = o1[i] * inv;
    dst[32 + n] = o2[i] * inv;
    dst[48 + n] = o3[i] * inv;
  }
}

// ---------------------------------------------------------------------------
extern "C" void kernel_launch(void* const* d_in, const int* in_sizes, int n_in,
                              void* d_out, int out_size, void* d_ws, size_t ws_size,
                              hipStream_t stream) {
  const float* hs     = (const float*)d_in[0];
  const float* w_attn = (const float*)d_in[1];
  const float* b_attn = (const float*)d_in[2];
  const float* w_proj = (const float*)d_in[3];
  const float* b_proj = (const float*)d_in[4];
  float* out = (float*)d_out;

  const size_t chunk = (size_t)BS * Dm;      // 4 Mi floats = 16 MB
  float* ws = (float*)d_ws;
  float* qw = ws;
  float* kw = ws + chunk;
  float* vw = ws + 2 * chunk;
  float* aw = ws + 3 * chunk;

  // 1) QKV projection -> head-layout q/k/v
  dim3 g1(3 * Dm / 64, BS / 16);             // 48 x 256
  gemm_kernel<3 * Dm, 0><<<g1, 128, 0, stream>>>(hs, w_attn, b_attn, qw, kw, vw);

  // 2) causal attention -> [B,S,D] workspace
  dim3 g2(Sq / 64, Bb * Hh);                 // 32 x 32
  attn_kernel<<<g2, 128, 0, stream>>>(qw, kw, vw, aw);

  // 3) output projection -> d_out
  dim3 g3(Dm / 64, BS / 16);                 // 16 x 256
  gemm_kernel<Dm, 1><<<g3, 128, 0, stream>>>(aw, w_proj, b_proj, out, nullptr, nullptr);
}